// MPNEncoder_15530601742850
// MI455X (gfx1250) — compile-verified
//
#include <hip/hip_runtime.h>
#include <hip/hip_bf16.h>

// ---------------------------------------------------------------------------
// MPNN encoder for gfx1250 (MI455X).  fp32 WMMA (V_WMMA_F32_16X16X4_F32).
// Block tile 64(M) x 320(N), 8 wave32s arranged 4(M) x 2(N): 10 N-tiles /
// wave (80 accumulator VGPRs).  B is staged in LDS k-quad interleaved
// ( [(k>>2)][col][k&3] ) so each WMMA B operand pair {k,k+1} is one aligned
// ds_load_b64 straight into an even VGPR pair: no repacking movs, and the
// two half-waves (koff 0/2) touch disjoint bank quads -> conflict-free.
// ---------------------------------------------------------------------------

typedef float v2f __attribute__((ext_vector_type(2)));
typedef float v8f __attribute__((ext_vector_type(8)));

#define HID     300     // hidden
#define NPAD    320     // hidden padded to 20 WMMA N-tiles (2 halves x 10)
#define NTW     10      // N tiles of 16 per wave
#define BM      64      // rows per block (4 M-waves x 16)
#define KC      32      // K chunk staged in LDS (8 WMMA k-steps of 4)
#define APITCH  36      // As row pitch (even: keeps {k,k+1} 8B aligned)
#define TPB     256
#define N_ATOMS 100000
#define N_BONDS 200000
#define N_MOLS  4000
#define AFD     133     // atom feature dim
#define BFD     147     // bond feature dim
#define MAX_NB  6

// ----------------------------- WMMA GEMM -----------------------------------
// MODE 0: inp = f_bonds @ W_i           ; out0=inp, out1=relu(inp)
// MODE 1: msgOut = relu(inp + (amsg[b2a]-msgIn[b2revb]) @ W_h)
// MODE 2: h = relu([f_atoms|amsg] @ W_o + b) ; atomicAdd into mol sums (d_out)
template <int MODE>
__global__ __launch_bounds__(TPB)
void mpn_gemm(const float* __restrict__ Asrc0,   // M0:f_bonds  M1:amsg    M2:f_atoms
              const float* __restrict__ Asrc1,   // M1:msgIn    M2:amsg
              const int*   __restrict__ idx0,    // M1:b2a      M2:atom2mol
              const int*   __restrict__ idx1,    // M1:b2revb
              const float* __restrict__ W,       // [K x 300] row-major
              const float* __restrict__ bias,    // M2: W_o_b
              const float* __restrict__ inp,     // M1: inp
              float* __restrict__ out0,
              float* __restrict__ out1,
              int M, int K)
{
    __shared__ float As[BM * APITCH];            // [m][k], pitch 36
    __shared__ float Bs[(KC / 4) * NPAD * 4];    // [(k>>2)][col][k&3]

    const int t    = threadIdx.x;
    const int lane = t & 31;
    const int wave = t >> 5;
    const int wm   = wave & 3;           // M-slice of this wave (0..3)
    const int wn   = wave >> 2;          // N-half of this wave (0..1)
    const int r0   = blockIdx.x * BM;
    const int n    = lane & 15;          // N within 16-tile / M within 16 for A
    const int koff = (lane >> 4) << 1;   // upper half-wave handles K+2,K+3
    const int nbase = wn * (NTW * 16);   // first column of this wave's N-half

    v8f acc[NTW];
    #pragma unroll
    for (int i = 0; i < NTW; ++i) acc[i] = v8f{0,0,0,0,0,0,0,0};

    const int numKC = (K + KC - 1) / KC;
    for (int kc = 0; kc < numKC; ++kc) {
        const int k0 = kc * KC;
        __syncthreads();
        // ---- stage B chunk [KC x NPAD], zero-padded, k-quad interleaved ---
        #pragma unroll
        for (int i = 0; i < (KC * NPAD) / TPB; ++i) {       // 40 elems/thread
            int li = t + i * TPB;
            int br = li / NPAD;          // k within chunk
            int bc = li - br * NPAD;     // column
            int gk = k0 + br;
            float v = 0.0f;
            if (gk < K && bc < HID) v = W[gk * HID + bc];
            Bs[((br >> 2) * NPAD + bc) * 4 + (br & 3)] = v;
        }
        // ---- stage A chunk [BM x KC] (gathers fused here) -----------------
        #pragma unroll
        for (int i = 0; i < (BM * KC) / TPB; ++i) {         // 8 elems/thread
            int li = t + i * TPB;
            int ar = li >> 5;            // / KC
            int ac = li & (KC - 1);
            int gr = r0 + ar;
            int gk = k0 + ac;
            float v = 0.0f;
            if (gr < M && gk < K) {
                if (MODE == 0) {
                    v = Asrc0[gr * BFD + gk];
                } else if (MODE == 1) {
                    int ba = idx0[gr];   // b2a
                    int rb = idx1[gr];   // b2revb
                    v = Asrc0[ba * HID + gk] - Asrc1[rb * HID + gk];
                } else {
                    v = (gk < AFD) ? Asrc0[gr * AFD + gk]
                                   : Asrc1[gr * HID + (gk - AFD)];
                }
            }
            As[ar * APITCH + ac] = v;
        }
        __syncthreads();
        // ---- 8 k-steps of 4, 10 N-tiles each ------------------------------
        const int ml = wm * 16 + n;      // A row owned by this lane
        #pragma unroll
        for (int ks = 0; ks < KC; ks += 4) {
            // {A[ml][ks+koff], A[ml][ks+koff+1]}: contiguous, 8B aligned
            v2f a = *(const v2f*)&As[ml * APITCH + ks + koff];
            const int brow = (ks >> 2) * (NPAD * 4);
            #pragma unroll
            for (int tt = 0; tt < NTW; ++tt) {
                int c = nbase + tt * 16 + n;
                // {B[ks+koff][c], B[ks+koff+1][c]}: one aligned b64 load
                v2f b = *(const v2f*)&Bs[brow + c * 4 + koff];
                acc[tt] = __builtin_amdgcn_wmma_f32_16x16x4_f32(
                    false, a, false, b, (short)0, acc[tt], false, false);
            }
        }
    }
    // ---- epilogue: C/D layout -> VGPR j holds M=j (lanes 0-15), M=j+8 -----
    const int mbase = r0 + wm * 16 + (lane >> 4) * 8;
    #pragma unroll
    for (int tt = 0; tt < NTW; ++tt) {
        int col = nbase + tt * 16 + n;
        if (col >= HID) continue;
        #pragma unroll
        for (int j = 0; j < 8; ++j) {
            int gr = mbase + j;
            if (gr < M) {
                float v = acc[tt][j];
                if (MODE == 0) {
                    out0[gr * HID + col] = v;
                    out1[gr * HID + col] = fmaxf(v, 0.0f);
                } else if (MODE == 1) {
                    out0[gr * HID + col] =
                        fmaxf(inp[gr * HID + col] + v, 0.0f);
                } else {
                    float h = fmaxf(v + bias[col], 0.0f);
                    atomicAdd(&out0[idx0[gr] * HID + col], h);  // idx0=atom2mol
                }
            }
        }
    }
}

// ------------------- neighbor gather-sum: amsg = sum msg[a2b] ---------------
__global__ void gather_sum(const float* __restrict__ msg,
                           const int* __restrict__ a2b,
                           float* __restrict__ amsg, int n_atoms)
{
    int e = blockIdx.x * blockDim.x + threadIdx.x;
    int total = n_atoms * HID;
    if (e >= total) return;
    int a = e / HID;
    int c = e - a * HID;
    const int* nb = a2b + a * MAX_NB;
    float s = 0.0f;
    #pragma unroll
    for (int i = 0; i < MAX_NB; ++i) s += msg[nb[i] * HID + c];
    amsg[e] = s;
}

// ------------------------------- helpers ------------------------------------
__global__ void zero_out(float* __restrict__ out, float* __restrict__ counts)
{
    int i = blockIdx.x * blockDim.x + threadIdx.x;
    if (i < N_MOLS * HID) out[i] = 0.0f;
    if (i < N_MOLS) counts[i] = 0.0f;
}

__global__ void count_atoms(const int* __restrict__ atom2mol,
                            float* __restrict__ counts, int n)
{
    int i = blockIdx.x * blockDim.x + threadIdx.x;
    if (i < n) atomicAdd(&counts[atom2mol[i]], 1.0f);
}

__global__ void finalize_mean(float* __restrict__ out,
                              const float* __restrict__ counts)
{
    int i = blockIdx.x * blockDim.x + threadIdx.x;
    if (i < N_MOLS * HID) out[i] = out[i] / fmaxf(counts[i / HID], 1.0f);
}

// ------------------------------- driver -------------------------------------
extern "C" void kernel_launch(void* const* d_in, const int* in_sizes, int n_in,
                              void* d_out, int out_size, void* d_ws, size_t ws_size,
                              hipStream_t stream)
{
    const float* f_atoms  = (const float*)d_in[0];
    const float* f_bonds  = (const float*)d_in[1];
    const int*   a2b      = (const int*)d_in[2];
    const int*   b2a      = (const int*)d_in[3];
    const int*   b2revb   = (const int*)d_in[4];
    const int*   atom2mol = (const int*)d_in[5];
    const float* W_i      = (const float*)d_in[6];
    const float* W_h      = (const float*)d_in[7];
    const float* W_o_w    = (const float*)d_in[8];
    const float* W_o_b    = (const float*)d_in[9];
    float* out = (float*)d_out;

    // workspace layout (floats)
    float* inp    = (float*)d_ws;                    // 200000*300
    float* msgA   = inp  + (size_t)N_BONDS * HID;    // 200000*300
    float* msgB   = msgA + (size_t)N_BONDS * HID;    // 200000*300
    float* amsg   = msgB + (size_t)N_BONDS * HID;    // 100000*300
    float* counts = amsg + (size_t)N_ATOMS * HID;    // 4000

    const int gemm_bonds = (N_BONDS + BM - 1) / BM;  // 3125
    const int gemm_atoms = (N_ATOMS + BM - 1) / BM;  // 1563
    const int gs_blocks  = (N_ATOMS * HID + TPB - 1) / TPB;
    const int mol_blocks = (N_MOLS * HID + TPB - 1) / TPB;

    // mol sums (d_out) and counts must start at zero
    zero_out<<<mol_blocks, TPB, 0, stream>>>(out, counts);
    count_atoms<<<(N_ATOMS + TPB - 1) / TPB, TPB, 0, stream>>>(atom2mol, counts, N_ATOMS);

    // inp = f_bonds @ W_i ; msgA = relu(inp)
    mpn_gemm<0><<<gemm_bonds, TPB, 0, stream>>>(
        f_bonds, nullptr, nullptr, nullptr, W_i, nullptr, nullptr,
        inp, msgA, N_BONDS, BFD);

    // depth iteration 1
    gather_sum<<<gs_blocks, TPB, 0, stream>>>(msgA, a2b, amsg, N_ATOMS);
    mpn_gemm<1><<<gemm_bonds, TPB, 0, stream>>>(
        amsg, msgA, b2a, b2revb, W_h, nullptr, inp,
        msgB, nullptr, N_BONDS, HID);

    // depth iteration 2
    gather_sum<<<gs_blocks, TPB, 0, stream>>>(msgB, a2b, amsg, N_ATOMS);
    mpn_gemm<1><<<gemm_bonds, TPB, 0, stream>>>(
        amsg, msgB, b2a, b2revb, W_h, nullptr, inp,
        msgA, nullptr, N_BONDS, HID);

    // final atom aggregation + output GEMM fused with segment-sum
    gather_sum<<<gs_blocks, TPB, 0, stream>>>(msgA, a2b, amsg, N_ATOMS);
    mpn_gemm<2><<<gemm_atoms, TPB, 0, stream>>>(
        f_atoms, amsg, atom2mol, nullptr, W_o_w, W_o_b, nullptr,
        out, nullptr, N_ATOMS, AFD + HID);

    // mean = sum / max(count, 1)
    finalize_mean<<<mol_blocks, TPB, 0, stream>>>(out, counts);
}